// TemporalAttention_V2A_84241488544294
// MI455X (gfx1250) — compile-verified
//
#include <hip/hip_runtime.h>
#include <hip/hip_bf16.h>
#include <math.h>

// ---------------------------------------------------------------------------
// TemporalAttention V2A for MI455X (gfx1250, wave32).
//
// Cost model: output = passthrough copies of x1..x4 (~803MB) + 640x256 audio.
// HBM floor ~= 1.6GB traffic ~= 69us @ 23.3TB/s.  Branch compute ~8 GFLOP,
// done with V_WMMA_F32_16X16X4_F32 (exact f32) in a persistent per-(branch,
// 3-batch) workgroup kernel with all activations resident in LDS.
// ---------------------------------------------------------------------------

typedef float v2f __attribute__((ext_vector_type(2)));
typedef float v8f __attribute__((ext_vector_type(8)));

#define TT        5
#define DM        256
#define DI        512
#define DSTATE    16
#define DTRANK    16
#define NBR       4
#define NLAY      3
#define BB        128          // batch
#define BT        640          // BB*TT

// LDS strides (floats). Chosen so (stride % 64) is a small non-zero multiple
// of 4 -> A-fragment column reads across 16 rows hit 16 distinct banks.
#define SM  260   // for 256-wide buffers
#define SI  516   // for 512-wide buffers
#define SD  52    // for 48-wide dbl buffer

#define SX_OFF    0
#define STMP_OFF  (16*SM)                 // 4160
#define SXC_OFF   (STMP_OFF + 16*SI)      // +8256
#define SZ_OFF    (SXC_OFF  + 16*SI)
#define SDBL_OFF  (SZ_OFF   + 16*SI)
#define SMEM_FLOATS (SDBL_OFF + 16*SD)    // 29760 floats
#define SMEM_BYTES  (SMEM_FLOATS * 4)     // 119040 B (< 320KB/WGP)

// ---- d_out layout (floats): [x4, x3, x2, x1, audio] ----
#define OFF_X4  0
#define OFF_X3  8028160
#define OFF_X2  40140800
#define OFF_X1  72253440
#define OFF_AU  200704000

__device__ __forceinline__ float silu(float v) { return v / (1.0f + __expf(-v)); }

// D = A(16xK from LDS, row-major w/ stride sstr) * B(Kx16, B[k][n]=W[n0+n][k])
// A frag: a.x=A[m][k0+ko], a.y=A[m][k0+ko+1], m=lane&15, ko=(lane>>4)*2
// B frag: b.x=W[n0+n][k0+ko], b.y=..+1,       n=lane&15
__device__ __forceinline__ v8f gemm_tile(const float* __restrict__ S, int sstr,
                                         const float* __restrict__ W, int ldw,
                                         int n0, int K, int lane) {
  const int idx = lane & 15;
  const int ko  = (lane >> 4) * 2;
  const float* srow = S + idx * sstr + ko;
  const float* wrow = W + (size_t)(n0 + idx) * ldw + ko;
  v8f c = {0.f,0.f,0.f,0.f,0.f,0.f,0.f,0.f};
  for (int k0 = 0; k0 < K; k0 += 4) {
    v2f a = *(const v2f*)(srow + k0);
    v2f b = *(const v2f*)(wrow + k0);
    c = __builtin_amdgcn_wmma_f32_16x16x4_f32(false, a, false, b,
                                              (short)0, c, false, false);
  }
  return c;
}

// ---------------------------------------------------------------------------
// Kernel 1: zero gate-sum accumulator
// ---------------------------------------------------------------------------
__global__ void zero_kernel(float* __restrict__ p, int n) {
  int i = blockIdx.x * blockDim.x + threadIdx.x;
  if (i < n) p[i] = 0.0f;
}

// ---------------------------------------------------------------------------
// Kernel 2: fused passthrough-copy + spatial mean.  One wave per (bt,c) row.
// src: (640,256,H,W) f32.  dst: same shape slice of d_out.  xsd[row]=mean.
// ---------------------------------------------------------------------------
__global__ __launch_bounds__(256)
void copy_mean_kernel(const float* __restrict__ src, float* __restrict__ dst,
                      float* __restrict__ xsd, int HW, int hw4) {
  const int lane = threadIdx.x & 31;
  const int wid  = threadIdx.x >> 5;
  const size_t row = (size_t)blockIdx.x * 8 + wid;   // < 163840
  float sum = 0.f;
  if (hw4) {
    const float4* s4 = (const float4*)src;
    float4* d4 = (float4*)dst;
    const size_t base = row * (size_t)hw4;
    for (int e = lane; e < hw4; e += 32) {
      float4 v = s4[base + e];
      d4[base + e] = v;
      sum += v.x + v.y + v.z + v.w;
    }
  } else {
    const size_t base = row * (size_t)HW;
    for (int e = lane; e < HW; e += 32) {
      float v = src[base + e];
      dst[base + e] = v;
      sum += v;
    }
  }
  #pragma unroll
  for (int off = 16; off > 0; off >>= 1) sum += __shfl_xor(sum, off, 32);
  if (lane == 0) xsd[row] = sum * (1.0f / (float)HW);
}

// ---------------------------------------------------------------------------
// Kernel 3: full branch pipeline.  grid = (43, 4): (3-batch tile, branch).
// 256 threads = 8 waves.  rows = nb*5 <= 15 padded to the 16-row WMMA M-tile
// (rows >= `rows` compute garbage that is never stored).
// ---------------------------------------------------------------------------
__global__ __launch_bounds__(256, 1)
void branch_kernel(const float* __restrict__ xs,
                   const float* __restrict__ vfc_W, const float* __restrict__ vfc_b,
                   const float* __restrict__ ln_w,  const float* __restrict__ ln_b,
                   const float* __restrict__ in_W,  const float* __restrict__ conv_w,
                   const float* __restrict__ conv_b,const float* __restrict__ xproj_W,
                   const float* __restrict__ dt_W,  const float* __restrict__ dt_b,
                   const float* __restrict__ Alog,  const float* __restrict__ Dp,
                   const float* __restrict__ out_W, const float* __restrict__ gate_W,
                   const float* __restrict__ gate_b, float* __restrict__ gsum) {
  extern __shared__ float smem[];
  float* sX   = smem + SX_OFF;    // residual stream, 16 x SM
  float* sTMP = smem + STMP_OFF;  // stage / xc_raw / delta, 16 x SI
  float* sXC  = smem + SXC_OFF;   // conv+silu, 16 x SI
  float* sZ   = smem + SZ_OFF;    // silu(z) then y*silu(z), 16 x SI
  float* sDBL = smem + SDBL_OFF;  // dt|B|C, 16 x SD

  const int br   = blockIdx.y;
  const int b0   = blockIdx.x * 3;
  const int nb   = min(3, BB - b0);
  const int rows = nb * TT;
  const int tid  = threadIdx.x;
  const int lane = tid & 31;
  const int wid  = tid >> 5;
  const int tn   = lane & 15;          // D-frag col
  const int tmb  = (lane >> 4) * 8;    // D-frag row base

  // ---- stage xs rows, then vfc GEMM: sX = xs @ vfc_W^T + vfc_b ----
  for (int e = tid; e < rows * DM; e += 256) {
    int r = e >> 8, cix = e & 255;
    sTMP[r * SM + cix] = xs[((size_t)br * BT + (size_t)b0 * TT + r) * DM + cix];
  }
  __syncthreads();
  for (int t = 0; t < 2; ++t) {
    int n0 = (wid * 2 + t) * 16;
    v8f c = gemm_tile(sTMP, SM, vfc_W + (size_t)br * DM * DM, DM, n0, DM, lane);
    #pragma unroll
    for (int i = 0; i < 8; ++i) {
      int m = tmb + i;
      if (m < rows) sX[m * SM + n0 + tn] = c[i] + vfc_b[br * DM + n0 + tn];
    }
  }
  __syncthreads();

  for (int L = 0; L < NLAY; ++L) {
    const size_t lo = (size_t)(br * NLAY + L);

    // ---- LayerNorm, in place on sX (wave per row) ----
    for (int r = wid; r < rows; r += 8) {
      float s1 = 0.f, s2 = 0.f;
      for (int cix = lane; cix < DM; cix += 32) {
        float v = sX[r * SM + cix]; s1 += v; s2 += v * v;
      }
      #pragma unroll
      for (int off = 16; off > 0; off >>= 1) {
        s1 += __shfl_xor(s1, off, 32); s2 += __shfl_xor(s2, off, 32);
      }
      float mean = s1 * (1.0f / DM);
      float rstd = rsqrtf(s2 * (1.0f / DM) - mean * mean + 1e-5f);
      for (int cix = lane; cix < DM; cix += 32) {
        float v = sX[r * SM + cix];
        sX[r * SM + cix] = (v - mean) * rstd * ln_w[lo * DM + cix] + ln_b[lo * DM + cix];
      }
    }
    __syncthreads();

    // ---- in-proj: xc_raw -> sTMP, silu(z) -> sZ ----
    const float* inw = in_W + lo * (2 * DI) * DM;
    for (int t = 0; t < 4; ++t) {
      int n0 = (wid * 4 + t) * 16;
      v8f c = gemm_tile(sX, SM, inw, DM, n0, DM, lane);
      #pragma unroll
      for (int i = 0; i < 8; ++i) {
        int m = tmb + i;
        if (m < rows) sTMP[m * SI + n0 + tn] = c[i];
      }
    }
    for (int t = 0; t < 4; ++t) {
      int n0 = (wid * 4 + t) * 16;
      v8f c = gemm_tile(sX, SM, inw + (size_t)DI * DM, DM, n0, DM, lane);
      #pragma unroll
      for (int i = 0; i < 8; ++i) {
        int m = tmb + i;
        if (m < rows) sZ[m * SI + n0 + tn] = silu(c[i]);
      }
    }
    __syncthreads();

    // ---- depthwise causal conv (k=4, left pad 3) + silu -> sXC ----
    for (int e = tid; e < rows * DI; e += 256) {
      int r = e >> 9, d = e & 511;
      int lb = r / TT, tt = r - lb * TT;
      const float* cw = conv_w + (lo * DI + d) * 4;
      float acc = conv_b[lo * DI + d];
      #pragma unroll
      for (int k = 0; k < 4; ++k) {
        int ts = tt - 3 + k;
        if (ts >= 0) acc += cw[k] * sTMP[(lb * TT + ts) * SI + d];
      }
      sXC[r * SI + d] = silu(acc);
    }
    __syncthreads();

    // ---- x-proj: dbl(48) = sXC @ xproj_W^T (waves 0..2) ----
    if (wid < 3) {
      int n0 = wid * 16;
      v8f c = gemm_tile(sXC, SI, xproj_W + lo * (DTRANK + 2 * DSTATE) * DI, DI, n0, DI, lane);
      #pragma unroll
      for (int i = 0; i < 8; ++i) {
        int m = tmb + i;
        if (m < rows) sDBL[m * SD + n0 + tn] = c[i];
      }
    }
    __syncthreads();

    // ---- delta = softplus(dt @ dt_W^T + dt_b) -> sTMP ----
    const float* dtw = dt_W + lo * DI * DTRANK;
    for (int t = 0; t < 4; ++t) {
      int n0 = (wid * 4 + t) * 16;
      v8f c = gemm_tile(sDBL, SD, dtw, DTRANK, n0, DTRANK, lane);
      #pragma unroll
      for (int i = 0; i < 8; ++i) {
        int m = tmb + i;
        if (m < rows) {
          float v = c[i] + dt_b[lo * DI + n0 + tn];
          sTMP[m * SI + n0 + tn] = (v > 20.f) ? v : log1pf(__expf(v));
        }
      }
    }
    __syncthreads();

    // ---- selective scan over T, then yz = y*silu(z) in place in sZ ----
    for (int e = tid; e < nb * DI; e += 256) {
      int lb = e >> 9, d = e & 511;
      const float* alog = Alog + (lo * DI + d) * DSTATE;
      float As[DSTATE], h[DSTATE];
      #pragma unroll
      for (int s = 0; s < DSTATE; ++s) { As[s] = -__expf(alog[s]); h[s] = 0.f; }
      const float Dv = Dp[lo * DI + d];
      for (int tt = 0; tt < TT; ++tt) {
        int r = lb * TT + tt;
        float delta = sTMP[r * SI + d];
        float xc    = sXC[r * SI + d];
        const float* dblr = sDBL + r * SD;
        float acc = 0.f;
        #pragma unroll
        for (int s = 0; s < DSTATE; ++s) {
          float dA = __expf(delta * As[s]);
          h[s] = dA * h[s] + delta * dblr[DTRANK + s] * xc;
          acc += h[s] * dblr[DTRANK + DSTATE + s];
        }
        float y = acc + Dv * xc;
        sZ[r * SI + d] = y * sZ[r * SI + d];
      }
    }
    __syncthreads();

    // ---- out-proj + residual: sX += yz @ out_W^T ----
    const float* ow = out_W + lo * DM * DI;
    for (int t = 0; t < 2; ++t) {
      int n0 = (wid * 2 + t) * 16;
      v8f c = gemm_tile(sZ, SI, ow, DI, n0, DI, lane);
      #pragma unroll
      for (int i = 0; i < 8; ++i) {
        int m = tmb + i;
        if (m < rows) sX[m * SM + n0 + tn] += c[i];
      }
    }
    __syncthreads();
  }

  // ---- gate = silu(sX @ gate_W^T + gate_b); accumulate mean transposed ----
  for (int t = 0; t < 2; ++t) {
    int n0 = (wid * 2 + t) * 16;
    v8f c = gemm_tile(sX, SM, gate_W + (size_t)br * DM * DM, DM, n0, DM, lane);
    #pragma unroll
    for (int i = 0; i < 8; ++i) {
      int m = tmb + i;
      if (m < rows) {
        float v = silu(c[i] + gate_b[br * DM + n0 + tn]);
        int lb = m / TT, tt = m - lb * TT;
        atomicAdd(&gsum[((size_t)tt * BB + b0 + lb) * DM + n0 + tn], v);
      }
    }
  }
}

// ---------------------------------------------------------------------------
// Kernel 4: audio_out[b*T+t, c] = a * (1 + mean_gate[t*B+b, c])
// ---------------------------------------------------------------------------
__global__ __launch_bounds__(256)
void audio_kernel(const float* __restrict__ audio, const float* __restrict__ gsum,
                  float* __restrict__ out) {
  int i = blockIdx.x;            // 0..639 : b*T + t
  int c = threadIdx.x;           // 0..255
  int b = i / TT, t = i - b * TT;
  float a = audio[i * DM + c];
  float g = 0.25f * gsum[((size_t)t * BB + b) * DM + c];
  out[i * DM + c] = a + g * a;
}

// ---------------------------------------------------------------------------
extern "C" void kernel_launch(void* const* d_in, const int* in_sizes, int n_in,
                              void* d_out, int out_size, void* d_ws, size_t ws_size,
                              hipStream_t stream) {
  (void)in_sizes; (void)n_in; (void)out_size; (void)ws_size;
  const float* x1      = (const float*)d_in[0];
  const float* x2      = (const float*)d_in[1];
  const float* x3      = (const float*)d_in[2];
  const float* x4      = (const float*)d_in[3];
  const float* audio   = (const float*)d_in[4];
  const float* vfc_W   = (const float*)d_in[5];
  const float* vfc_b   = (const float*)d_in[6];
  const float* ln_w    = (const float*)d_in[7];
  const float* ln_b    = (const float*)d_in[8];
  const float* in_W    = (const float*)d_in[9];
  const float* conv_w  = (const float*)d_in[10];
  const float* conv_b  = (const float*)d_in[11];
  const float* xproj_W = (const float*)d_in[12];
  const float* dt_W    = (const float*)d_in[13];
  const float* dt_b    = (const float*)d_in[14];
  const float* Alog    = (const float*)d_in[15];
  const float* Dpp     = (const float*)d_in[16];
  const float* out_W   = (const float*)d_in[17];
  const float* gate_W  = (const float*)d_in[18];
  const float* gate_b  = (const float*)d_in[19];
  float* out = (float*)d_out;

  float* xs   = (float*)d_ws;               // 4 * 640 * 256
  float* gsum = xs + (size_t)NBR * BT * DM; // 640 * 256

  static bool attr_set = false;
  if (!attr_set) {   // host-side attribute, not a stream op (capture-safe)
    hipFuncSetAttribute((const void*)branch_kernel,
                        hipFuncAttributeMaxDynamicSharedMemorySize, SMEM_BYTES);
    attr_set = true;
  }

  zero_kernel<<<(BT * DM + 255) / 256, 256, 0, stream>>>(gsum, BT * DM);

  const int nrows = BT * DM;                 // 163840 rows per input
  const int nblk  = nrows / 8;               // 8 waves (rows) per block
  copy_mean_kernel<<<nblk, 256, 0, stream>>>(x1, out + OFF_X1, xs + 0 * (size_t)nrows, 784, 196);
  copy_mean_kernel<<<nblk, 256, 0, stream>>>(x2, out + OFF_X2, xs + 1 * (size_t)nrows, 196, 49);
  copy_mean_kernel<<<nblk, 256, 0, stream>>>(x3, out + OFF_X3, xs + 2 * (size_t)nrows, 196, 49);
  copy_mean_kernel<<<nblk, 256, 0, stream>>>(x4, out + OFF_X4, xs + 3 * (size_t)nrows, 49, 0);

  branch_kernel<<<dim3(43, 4), 256, SMEM_BYTES, stream>>>(
      xs, vfc_W, vfc_b, ln_w, ln_b, in_W, conv_w, conv_b, xproj_W,
      dt_W, dt_b, Alog, Dpp, out_W, gate_W, gate_b, gsum);

  audio_kernel<<<BT, 256, 0, stream>>>(audio, gsum, out + OFF_AU);
}